// SAGEModel_30434138259919
// MI455X (gfx1250) — compile-verified
//
#include <hip/hip_runtime.h>
#include <hip/hip_bf16.h>

// ---------------------------------------------------------------------------
// GraphSAGE (2x SAGEConv mean + classifier) for MI455X / gfx1250, wave32.
//   h0 = relu(key_emb[f0] + val_emb[f1])                       [N,128]
//   agg = segment_mean(h[src] -> dst)                          [N,128]
//   h'  = h @ Wself^T + agg @ Wneigh^T        (fused f32 WMMA GEMM)
//   out = h2 @ Wcls^T                                          [N,64]
// Edge scatter dominates bandwidth; GEMMs use V_WMMA_F32_16X16X4_F32.
// ---------------------------------------------------------------------------

#define N_NODES 50000   // multiple of 16 -> no GEMM bounds checks, EXEC all-1s
#define N_EDGES 800000
#define DIM     128
#define OUTD    64

typedef __attribute__((ext_vector_type(2))) float v2f;
typedef __attribute__((ext_vector_type(8))) float v8f;

// --- embedding sum + ReLU ---------------------------------------------------
__global__ __launch_bounds__(256)
void embed_relu_kernel(const int* __restrict__ feats,
                       const float* __restrict__ key_emb,
                       const float* __restrict__ val_emb,
                       float* __restrict__ h) {
  int tid = blockIdx.x * blockDim.x + threadIdx.x;   // N*32 threads, float4 each
  int n = tid >> 5;
  if (n >= N_NODES) return;
  int c = (tid & 31) << 2;
  int f0 = feats[n * 2 + 0];
  int f1 = feats[n * 2 + 1];
  float4 k4 = *(const float4*)(key_emb + (size_t)f0 * DIM + c);
  float4 v4 = *(const float4*)(val_emb + (size_t)f1 * DIM + c);
  float4 r;
  r.x = fmaxf(k4.x + v4.x, 0.0f);
  r.y = fmaxf(k4.y + v4.y, 0.0f);
  r.z = fmaxf(k4.z + v4.z, 0.0f);
  r.w = fmaxf(k4.w + v4.w, 0.0f);
  *(float4*)(h + (size_t)n * DIM + c) = r;
}

// --- in-degree (float, shared by both layers) -------------------------------
__global__ __launch_bounds__(256)
void degree_kernel(const int* __restrict__ edge_index,  // [2,E]
                   float* __restrict__ deg) {
  int e = blockIdx.x * blockDim.x + threadIdx.x;
  if (e >= N_EDGES) return;
  atomicAdd(&deg[edge_index[N_EDGES + e]], 1.0f);
}

// --- edge gather + scatter-add: agg[dst] += h[src] --------------------------
// 32 lanes per edge: coalesced 512B row read, 4 f32 atomics per lane.
__global__ __launch_bounds__(256)
void scatter_add_kernel(const int* __restrict__ edge_index,  // [2,E]
                        const float* __restrict__ h,
                        float* __restrict__ agg) {
  long long tid = (long long)blockIdx.x * blockDim.x + threadIdx.x;
  int e = (int)(tid >> 5);
  if (e >= N_EDGES) return;
  int c = ((int)tid & 31) << 2;
  int s = edge_index[e];
  int d = edge_index[N_EDGES + e];
  float4 v = *(const float4*)(h + (size_t)s * DIM + c);
  float* a = agg + (size_t)d * DIM + c;
  atomicAdd(a + 0, v.x);
  atomicAdd(a + 1, v.y);
  atomicAdd(a + 2, v.z);
  atomicAdd(a + 3, v.w);
}

// --- fused SAGE GEMM via V_WMMA_F32_16X16X4_F32 -----------------------------
// O[mrow:mrow+16, :CO] = A1 @ W1^T (+ (A2/max(deg,1)) @ W2^T)
// One wave per 16x16 output tile; block = CO/16 waves.
// A-lane: m = lane%16, k_base = 2*(lane/16)  -> contiguous float2 of A row.
// B-lane: n = lane%16, same k_base           -> contiguous float2 of W[n][k]
//         (row-major W gives W^T operand for free).
// C/D:    c[r] holds (row = r + 8*(lane/16), col = lane%16).
template <int CO, bool FUSED>
__global__ __launch_bounds__(2 * CO)
void sage_gemm_kernel(const float* __restrict__ A1,
                      const float* __restrict__ W1,   // [CO, DIM]
                      const float* __restrict__ A2,
                      const float* __restrict__ W2,   // [CO, DIM]
                      const float* __restrict__ deg,
                      float* __restrict__ O) {        // [N, CO]
  const int lane = threadIdx.x & 31;
  const int wave = threadIdx.x >> 5;     // which 16-col tile
  const int m    = lane & 15;
  const int half = lane >> 4;
  const int kb   = half * 2;
  const int mrow = blockIdx.x * 16;
  const int ncol = wave * 16 + m;

  const float* a1p = A1 + (size_t)(mrow + m) * DIM + kb;
  const float* b1p = W1 + (size_t)ncol * DIM + kb;

  v8f c = {};
  if (FUSED) {
    const float* a2p = A2 + (size_t)(mrow + m) * DIM + kb;
    const float* b2p = W2 + (size_t)ncol * DIM + kb;
    const float inv = 1.0f / fmaxf(deg[mrow + m], 1.0f);
#pragma unroll
    for (int k = 0; k < DIM; k += 4) {
      v2f a1 = *(const v2f*)(a1p + k);
      v2f b1 = *(const v2f*)(b1p + k);
      c = __builtin_amdgcn_wmma_f32_16x16x4_f32(false, a1, false, b1,
                                                (short)0, c, false, false);
      v2f a2 = *(const v2f*)(a2p + k);
      a2 *= inv;                              // fold mean-normalization into A
      v2f b2 = *(const v2f*)(b2p + k);
      c = __builtin_amdgcn_wmma_f32_16x16x4_f32(false, a2, false, b2,
                                                (short)0, c, false, false);
    }
  } else {
#pragma unroll
    for (int k = 0; k < DIM; k += 4) {
      v2f a1 = *(const v2f*)(a1p + k);
      v2f b1 = *(const v2f*)(b1p + k);
      c = __builtin_amdgcn_wmma_f32_16x16x4_f32(false, a1, false, b1,
                                                (short)0, c, false, false);
    }
  }

  float* op = O + (size_t)(mrow + half * 8) * CO + wave * 16 + m;
#pragma unroll
  for (int r = 0; r < 8; ++r)
    op[(size_t)r * CO] = c[r];
}

// ---------------------------------------------------------------------------
extern "C" void kernel_launch(void* const* d_in, const int* in_sizes, int n_in,
                              void* d_out, int out_size, void* d_ws, size_t ws_size,
                              hipStream_t stream) {
  const int*   feats    = (const int*)d_in[0];   // [N,2]
  const int*   eidx     = (const int*)d_in[1];   // [2,E]
  const float* key_emb  = (const float*)d_in[2]; // [V,128]
  const float* val_emb  = (const float*)d_in[3];
  const float* Wself0   = (const float*)d_in[4]; // [128,128]
  const float* Wneigh0  = (const float*)d_in[5];
  const float* Wself1   = (const float*)d_in[6];
  const float* Wneigh1  = (const float*)d_in[7];
  const float* Wcls     = (const float*)d_in[8]; // [64,128]
  float* out = (float*)d_out;                    // [N,64]

  const size_t ND = (size_t)N_NODES * DIM;
  float* h0  = (float*)d_ws;          // [N,128]
  float* h1  = h0 + ND;               // [N,128]
  float* agg = h1 + ND;               // [N,128]
  float* deg = agg + ND;              // [N]

  // zero deg + agg (capture-safe memset nodes)
  hipMemsetAsync(agg, 0, ND * sizeof(float), stream);
  hipMemsetAsync(deg, 0, (size_t)N_NODES * sizeof(float), stream);

  const int embBlocks  = (N_NODES * 32 + 255) / 256;
  const int degBlocks  = (N_EDGES + 255) / 256;
  const int sctBlocks  = (int)(((long long)N_EDGES * 32 + 255) / 256);
  const int gemmBlocks = N_NODES / 16;   // 3125

  // h0 = relu(emb gather)
  embed_relu_kernel<<<embBlocks, 256, 0, stream>>>(feats, key_emb, val_emb, h0);
  degree_kernel<<<degBlocks, 256, 0, stream>>>(eidx, deg);

  // ---- layer 0 ----
  scatter_add_kernel<<<sctBlocks, 256, 0, stream>>>(eidx, h0, agg);
  sage_gemm_kernel<DIM, true><<<gemmBlocks, 256, 0, stream>>>(
      h0, Wself0, agg, Wneigh0, deg, h1);

  // ---- layer 1 ----
  hipMemsetAsync(agg, 0, ND * sizeof(float), stream);
  scatter_add_kernel<<<sctBlocks, 256, 0, stream>>>(eidx, h1, agg);
  sage_gemm_kernel<DIM, true><<<gemmBlocks, 256, 0, stream>>>(
      h1, Wself1, agg, Wneigh1, deg, h0);   // reuse h0 as h2

  // ---- classifier: out = h2 @ Wcls^T ----
  sage_gemm_kernel<OUTD, false><<<gemmBlocks, 128, 0, stream>>>(
      h0, Wcls, nullptr, nullptr, nullptr, out);
}